// KGWDiscriminator_79190607003920
// MI455X (gfx1250) — compile-verified
//
#include <hip/hip_runtime.h>
#include <hip/hip_bf16.h>
#include <stdint.h>

#define KGW_P      2147483647u   // 2^31 - 1 (Mersenne prime)
#define KGW_VOCAB  50257u
#define KGW_GREEN  12564u        // int(0.25 * 50257)
#define KGW_HASH   15485863u     // HASH_KEY % P == HASH_KEY
#define KGW_CW     4
#define KGW_SMAX   8192
#define KGW_BLOCK  256

typedef int v4i __attribute__((ext_vector_type(4)));
typedef __attribute__((address_space(3))) v4i as3_v4i;

// x mod (2^31-1) for x < 2^62 via Mersenne folding (exact).
__device__ __forceinline__ unsigned int kgw_mod_p(unsigned long long x) {
  x = (x & 0x7FFFFFFFull) + (x >> 31);  // < 2^31 + 2^31
  x = (x & 0x7FFFFFFFull) + (x >> 31);  // <= 2^31
  unsigned int r = (unsigned int)x;
  if (r >= KGW_P) r -= KGW_P;
  return r;
}

__global__ __launch_bounds__(KGW_BLOCK) void kgw_zscore_kernel(
    const int* __restrict__ tokens, const int* __restrict__ lengths,
    float* __restrict__ out, int S) {
  __shared__ __align__(16) int s_tok[KGW_SMAX];
  __shared__ unsigned int s_count;

  const int b   = blockIdx.x;
  const int tid = threadIdx.x;
  int len = lengths[b];
  if (len > S) len = S;

  // Sequences shorter than CW+1 score 0 (uniform exit: no barrier hazard).
  if (len < KGW_CW + 1) {
    if (tid == 0) out[b] = 0.0f;
    return;
  }
  if (tid == 0) s_count = 0u;

  const int* row = tokens + (size_t)b * (size_t)S;

#if __has_builtin(__builtin_amdgcn_global_load_async_to_lds_b128)
  {
    // CDNA5 async DMA path: global -> LDS in 16B granules, tracked by ASYNCcnt.
    v4i* g     = (v4i*)row;                 // generic vector ptr (per builtin sig)
    as3_v4i* l = (as3_v4i*)s_tok;           // LDS vector ptr
    const int nvec = (len + 3) >> 2;        // ceil(len / 4) granules, <= S/4
    for (int i = tid; i < nvec; i += KGW_BLOCK)
      __builtin_amdgcn_global_load_async_to_lds_b128(g + i, l + i, 0, 0);
  #if __has_builtin(__builtin_amdgcn_s_wait_asynccnt)
    __builtin_amdgcn_s_wait_asynccnt(0);
  #else
    asm volatile("s_wait_asynccnt 0x0" ::: "memory");
  #endif
  }
#else
  // Fallback: plain coalesced loads into LDS.
  for (int i = tid; i < len; i += KGW_BLOCK) s_tok[i] = row[i];
#endif
  __syncthreads();

  // Quad-vectorized scoring: t0 = CW + 4*tid + 1024*k is a multiple of 4, so
  // two aligned ds_load_b128 fetch tokens [t0-4 .. t0+3]; that 8-token window
  // scores 4 tokens (each needs its 4 predecessors). 0.5 LDS loads/token.
  unsigned int cnt = 0u;
  const v4i* sv = (const v4i*)s_tok;
  for (int t0 = KGW_CW + 4 * tid; t0 < len; t0 += 4 * KGW_BLOCK) {
    const v4i a = sv[(t0 - KGW_CW) >> 2];   // tokens t0-4 .. t0-1
    const v4i c = sv[t0 >> 2];              // tokens t0   .. t0+3
    const unsigned int w[8] = {
        (unsigned int)(a.x + 1), (unsigned int)(a.y + 1),
        (unsigned int)(a.z + 1), (unsigned int)(a.w + 1),
        (unsigned int)(c.x + 1), (unsigned int)(c.y + 1),
        (unsigned int)(c.z + 1), (unsigned int)(c.w + 1)};
#pragma unroll
    for (int j = 0; j < 4; ++j) {
      unsigned int h = KGW_HASH;
      h = kgw_mod_p((unsigned long long)h * w[j]);      // prev_4
      h = kgw_mod_p((unsigned long long)h * w[j + 1]);  // prev_3
      h = kgw_mod_p((unsigned long long)h * w[j + 2]);  // prev_2
      h = kgw_mod_p((unsigned long long)h * w[j + 3]);  // prev_1
      const unsigned int g = kgw_mod_p((unsigned long long)h * w[j + 4]);
      const unsigned int green = ((g % KGW_VOCAB) < KGW_GREEN) ? 1u : 0u;
      const unsigned int valid = ((t0 + j) < len) ? 1u : 0u;
      cnt += green & valid;
    }
  }
  atomicAdd(&s_count, cnt);
  __syncthreads();

  if (tid == 0) {
    float T = (float)(len - KGW_CW);                  // >= 1 here
    float z = ((float)s_count - 0.25f * T) / sqrtf(0.1875f * T);
    out[b] = 1.0f / (1.0f + expf(-(z - 2.0f) * 0.5f));
  }
}

extern "C" void kernel_launch(void* const* d_in, const int* in_sizes, int n_in,
                              void* d_out, int out_size, void* d_ws, size_t ws_size,
                              hipStream_t stream) {
  (void)n_in; (void)out_size; (void)d_ws; (void)ws_size;
  const int* tokens  = (const int*)d_in[0];   // token_ids, flat (B*S)
  const int* lengths = (const int*)d_in[1];   // lengths, (B)
  float* out = (float*)d_out;                 // rewards, (B) float32
  const int B = in_sizes[1];
  const int S = in_sizes[0] / B;              // 8192, must be <= KGW_SMAX
  kgw_zscore_kernel<<<dim3(B), dim3(KGW_BLOCK), 0, stream>>>(tokens, lengths, out, S);
}